// PS_53798760349955
// MI455X (gfx1250) — compile-verified
//
#include <hip/hip_runtime.h>
#include <math.h>

#define PI_F  3.14159265358979f
#define PIH_F 1.57079632679490f

typedef __attribute__((ext_vector_type(2))) float v2f;
typedef __attribute__((ext_vector_type(8))) float v8f;

// ---------------------------------------------------------------------------
// small elementwise utilities
// ---------------------------------------------------------------------------
__global__ void k_setf(float* p, long n, float v){
  long i=(long)blockIdx.x*blockDim.x+threadIdx.x; if(i<n) p[i]=v;
}
__global__ void k_copyc(const float2* a, float2* b, long n){
  long i=(long)blockIdx.x*blockDim.x+threadIdx.x; if(i<n) b[i]=a[i];
}
__global__ void k_r2c(const float* a, float2* b, long n){
  long i=(long)blockIdx.x*blockDim.x+threadIdx.x; if(i<n) b[i]=make_float2(a[i],0.f);
}
__global__ void k_real(const float2* a, float* b, long n){
  long i=(long)blockIdx.x*blockDim.x+threadIdx.x; if(i<n) b[i]=a[i].x;
}
__global__ void k_mag(const float2* a, float* b, long n){
  long i=(long)blockIdx.x*blockDim.x+threadIdx.x; if(i<n){
    float2 v=a[i]; b[i]=sqrtf(v.x*v.x+v.y*v.y);
  }
}
__global__ void k_powspec(float2* d, long n){
  long i=(long)blockIdx.x*blockDim.x+threadIdx.x; if(i<n){
    float2 v=d[i]; d[i]=make_float2(v.x*v.x+v.y*v.y,0.f);
  }
}
__global__ void k_subs(float* x, long n, const float* s){
  long i=(long)blockIdx.x*blockDim.x+threadIdx.x; if(i<n) x[i]-= *s;
}

// ---------------------------------------------------------------------------
// shared-memory radix-2 FFT, one line per block (wave32 friendly, 8KB LDS)
// ---------------------------------------------------------------------------
__global__ void k_fft1d(float2* data, int n, int log2n, int axis, float dirsign, float scale){
  __shared__ float2 sh[1024];
  int line=blockIdx.x, tid=threadIdx.x, nt=blockDim.x;
  long base; long stride;
  if(axis==0){ base=(long)line*n; stride=1; } else { base=line; stride=n; }
  for(int j=tid;j<n;j+=nt){
    int r=(int)(__brev((unsigned)j)>>(32-log2n));
    sh[r]=data[base+(long)j*stride];
  }
  __syncthreads();
  for(int s=1;s<=log2n;s++){
    int m=1<<s, half=m>>1;
    float w0=dirsign*(-2.0f*PI_F)/(float)m;       // hoisted twiddle step
    for(int k=tid;k<(n>>1);k+=nt){
      int grp=k>>(s-1), pos=k&(half-1);
      int i0=grp*m+pos, i1=i0+half;
      float ang=w0*(float)pos;
      float sn,c; __sincosf(ang,&sn,&c);
      float2 u=sh[i0], v=sh[i1];
      float tr=v.x*c-v.y*sn, ti=v.x*sn+v.y*c;
      sh[i0]=make_float2(u.x+tr,u.y+ti);
      sh[i1]=make_float2(u.x-tr,u.y-ti);
    }
    __syncthreads();
  }
  for(int j=tid;j<n;j+=nt){
    float2 v=sh[j];
    data[base+(long)j*stride]=make_float2(v.x*scale,v.y*scale);
  }
}

// ---------------------------------------------------------------------------
// steerable-pyramid masks, evaluated analytically in UNSHIFTED freq coords
// (all sizes are powers of two: shift/mask indexing only, no int division)
// ---------------------------------------------------------------------------
static __device__ __forceinline__ float coordOf(int k, int n){
  int h=n>>1; int p=(k+h)&(n-1); return (float)(p-h)/(float)h;
}
static __device__ __forceinline__ float radOf(int kx,int ky,int n){
  float x=coordOf(kx,n), y=coordOf(ky,n);
  float r=sqrtf(x*x+y*y);
  if(kx==0&&ky==0) r=2.0f/(float)n;   // rad[h/2,w/2] := rad[h/2,w/2-1]
  return r;
}
__global__ void k_hi0lo0(const float2* dft, float2* hi, float2* lo, int n, int lg){
  long n2=(long)n*n; long i=(long)blockIdx.x*blockDim.x+threadIdx.x; if(i>=n2) return;
  int ky=(int)(i>>lg), kx=(int)i&(n-1);
  float lr=log2f(radOf(kx,ky,n));
  float t=fminf(fmaxf(lr+1.0f,0.f),1.f);
  float hm=__sinf(PIH_F*t), lm=__cosf(PIH_F*t);
  float2 d=dft[i];
  hi[i]=make_float2(d.x*hm,d.y*hm);
  lo[i]=make_float2(d.x*lm,d.y*lm);
}
__global__ void k_lo0m(float2* d, int n, int lg){  // multiply in place by lo0 mask
  long n2=(long)n*n; long i=(long)blockIdx.x*blockDim.x+threadIdx.x; if(i>=n2) return;
  int ky=(int)(i>>lg), kx=(int)i&(n-1);
  float t=fminf(fmaxf(log2f(radOf(kx,ky,n))+1.0f,0.f),1.f);
  float lm=__cosf(PIH_F*t);
  float2 v=d[i]; d[i]=make_float2(v.x*lm,v.y*lm);
}
__global__ void k_him(const float2* lod, float2* out, int n, int lg){
  long n2=(long)n*n; long i=(long)blockIdx.x*blockDim.x+threadIdx.x; if(i>=n2) return;
  int ky=(int)(i>>lg), kx=(int)i&(n-1);
  float t=fminf(fmaxf(log2f(radOf(kx,ky,n))+2.0f,0.f),1.f);
  float hm=__sinf(PIH_F*t);
  float2 v=lod[i]; out[i]=make_float2(v.x*hm,v.y*hm);
}
__global__ void k_bandmask(const float2* lod, float2* out, int n, float bang, int lg){
  long n2=(long)n*n; long i=(long)blockIdx.x*blockDim.x+threadIdx.x; if(i>=n2) return;
  int ky=(int)(i>>lg), kx=(int)i&(n-1);
  float x=coordOf(kx,n), y=coordOf(ky,n);
  float rad=(kx==0&&ky==0)?2.0f/(float)n:sqrtf(x*x+y*y);
  float lr=log2f(rad);
  float t2=fminf(fmaxf(lr+2.0f,0.f),1.f);
  float him=__sinf(PIH_F*t2);
  float th=atan2f(y,x)-bang;
  float u=fmodf(th+PI_F,2.0f*PI_F); if(u<0.f)u+=2.0f*PI_F;
  float thw=u-PI_F;
  float m=0.f;
  if(fabsf(thw)<PIH_F){ float c=__cosf(th); m=1.7888544f*c*c*c; } // 2*sqrt(0.8)*cos^3
  m*=him;
  float2 d=lod[i];
  out[i]=make_float2(-d.y*m, d.x*m);  // (-1j)^3 = i  ->  multiply by i
}
// crop shifted-center half + multiply by lo mask of the half grid
__global__ void k_croplo(const float2* in, float2* out, int n, int lgh){
  int h=n>>1; long n2=(long)h*h;
  long i=(long)blockIdx.x*blockDim.x+threadIdx.x; if(i>=n2) return;
  int oy=(int)(i>>lgh), ox=(int)i&(h-1);
  int opy=(oy+h/2)&(h-1), opx=(ox+h/2)&(h-1);
  float x=(float)(opx-h/2)/(float)(h/2), y=(float)(opy-h/2)/(float)(h/2);
  float rad=(ox==0&&oy==0)?2.0f/(float)h:sqrtf(x*x+y*y);
  float t=fminf(fmaxf(log2f(rad)+1.0f,0.f),1.f);
  float lm=__cosf(PIH_F*t);
  int iy=(opy+3*h/2)&(n-1), ix=(opx+3*h/2)&(n-1);
  float2 v=in[(long)iy*n+ix];
  out[i]=make_float2(v.x*lm,v.y*lm);
}
// zero-pad spectrum m2 -> m=2*m2 (out must be pre-zeroed)
__global__ void k_zeropad(const float2* in, float2* out, int m2, int lg2){
  int m=m2<<1; long n2=(long)m2*m2;
  long i=(long)blockIdx.x*blockDim.x+threadIdx.x; if(i>=n2) return;
  int ky=(int)(i>>lg2), kx=(int)i&(m2-1);
  int py=(ky+m2/2)&(m2-1), px=(kx+m2/2)&(m2-1);
  int oy=(py+3*m2/2)&(m-1), ox=(px+3*m2/2)&(m-1);
  out[(long)oy*m+ox]=in[i];
}

// ---------------------------------------------------------------------------
// parents / rparents construction
// ---------------------------------------------------------------------------
__global__ void k_parents(const float2* t, float* pm, float* pr, float* pi, long n2){
  long i=(long)blockIdx.x*blockDim.x+threadIdx.x; if(i>=n2) return;
  float r=t[i].x, im=t[i].y;
  float mag=sqrtf(r*r+im*im);
  pm[i]=mag;
  if(mag>1e-20f){ pr[i]=(im*im-r*r)/mag; pi[i]=2.f*r*im/mag; }
  else { pr[i]=0.f; pi[i]=0.f; }
}
__global__ void k_lowrp(const float2* t, float* p0,float* p1,float* p2,float* p3,float* p4,int m,int lg){
  long n2=(long)m*m; long i=(long)blockIdx.x*blockDim.x+threadIdx.x; if(i>=n2) return;
  int y=(int)(i>>lg), x=(int)i&(m-1);
  p0[i]=t[i].x*0.25f;
  p1[i]=t[(long)y*m+((x-1+m)&(m-1))].x*0.25f;
  p2[i]=t[(long)y*m+((x+1)&(m-1))].x*0.25f;
  p3[i]=t[(long)((y-1+m)&(m-1))*m+x].x*0.25f;
  p4[i]=t[(long)((y+1)&(m-1))*m+x].x*0.25f;
}

// ---------------------------------------------------------------------------
// reductions
// ---------------------------------------------------------------------------
__global__ void k_rsum(const float* x, long n, int pw, int useAbs, float scale, float* slot){
  __shared__ float sh[256];
  float acc=0.f;
  for(long i=(long)blockIdx.x*blockDim.x+threadIdx.x;i<n;i+=(long)gridDim.x*blockDim.x){
    float v=x[i]; if(useAbs) v=fabsf(v);
    float p=v; for(int q=1;q<pw;q++) p*=v;
    acc+=p;
  }
  sh[threadIdx.x]=acc; __syncthreads();
  for(int o=128;o>0;o>>=1){ if(threadIdx.x<o) sh[threadIdx.x]+=sh[threadIdx.x+o]; __syncthreads(); }
  if(threadIdx.x==0) atomicAdd(slot, sh[0]*scale);
}
static __device__ __forceinline__ void atomicMaxF(float* a, float v){
  unsigned* ua=(unsigned*)a; unsigned old=*ua;
  while(__uint_as_float(old)<v){ unsigned as=old; old=atomicCAS(ua,as,__float_as_uint(v)); if(old==as) break; }
}
static __device__ __forceinline__ void atomicMinF(float* a, float v){
  unsigned* ua=(unsigned*)a; unsigned old=*ua;
  while(__uint_as_float(old)>v){ unsigned as=old; old=atomicCAS(ua,as,__float_as_uint(v)); if(old==as) break; }
}
__global__ void k_minmax(const float* x, long n, float* mn, float* mx){
  __shared__ float smn[256], smx[256];
  float a=3.4e38f, b=-3.4e38f;
  for(long i=(long)blockIdx.x*blockDim.x+threadIdx.x;i<n;i+=(long)gridDim.x*blockDim.x){
    float v=x[i]; a=fminf(a,v); b=fmaxf(b,v);
  }
  smn[threadIdx.x]=a; smx[threadIdx.x]=b; __syncthreads();
  for(int o=128;o>0;o>>=1){
    if(threadIdx.x<o){ smn[threadIdx.x]=fminf(smn[threadIdx.x],smn[threadIdx.x+o]);
                       smx[threadIdx.x]=fmaxf(smx[threadIdx.x],smx[threadIdx.x+o]); }
    __syncthreads();
  }
  if(threadIdx.x==0){ atomicMinF(mn,smn[0]); atomicMaxF(mx,smx[0]); }
}
__global__ void k_central(const float* x, long n, const float* meanSlot, float* s2,float* s3,float* s4){
  __shared__ float a2[256],a3[256],a4[256];
  float mean=*meanSlot, c2=0.f,c3=0.f,c4=0.f;
  for(long i=(long)blockIdx.x*blockDim.x+threadIdx.x;i<n;i+=(long)gridDim.x*blockDim.x){
    float d=x[i]-mean, d2=d*d; c2+=d2; c3+=d2*d; c4+=d2*d2;
  }
  a2[threadIdx.x]=c2;a3[threadIdx.x]=c3;a4[threadIdx.x]=c4; __syncthreads();
  for(int o=128;o>0;o>>=1){
    if(threadIdx.x<o){ a2[threadIdx.x]+=a2[threadIdx.x+o]; a3[threadIdx.x]+=a3[threadIdx.x+o]; a4[threadIdx.x]+=a4[threadIdx.x+o]; }
    __syncthreads();
  }
  if(threadIdx.x==0){ atomicAdd(s2,a2[0]); atomicAdd(s3,a3[0]); atomicAdd(s4,a4[0]); }
}
__global__ void k_statg(float* scal, float* out, float n){
  if(threadIdx.x||blockIdx.x) return;
  float mean=scal[0];
  float var=scal[3]/(n-1.0f);
  out[0]=mean; out[1]=var;
  out[2]=(scal[4]/n)/(var*sqrtf(var));
  out[3]=(scal[5]/n)/(var*var);
  out[4]=scal[1]; out[5]=scal[2];
  scal[6]=var;                       // var0 for later "ok" tests
}
__global__ void k_skewkurt(const float* scal, float factor, float* so, float* ko){
  if(threadIdx.x||blockIdx.x) return;
  float vari=scal[8], var0=scal[6], m3=scal[9], m4=scal[10];
  int ok = (vari/var0)>1e-6f;
  *so = ok ? factor*m3/(vari*sqrtf(vari)) : 0.0f;
  *ko = ok ? factor*m4/(vari*vari) : 3.0f;
}
// write fftshifted 9x9 autocorr crop straight into d_out (stride-aware)
__global__ void k_acwrite(const float2* ac, int m, float invD, float* outBase, int strideIJ, float* variSlot){
  int idx=threadIdx.x; if(idx>=81) return;
  int dy=idx/9-4, dx=idx%9-4;
  float v=ac[(long)((dy+m)&(m-1))*m + ((dx+m)&(m-1))].x*invD;
  outBase[(long)idx*strideIJ]=v;
  if(dy==0&&dx==0&&variSlot) *variSlot=v;
}

// ---------------------------------------------------------------------------
// WMMA Gram / cross-correlation:  C(PxQ) = (1/denom) * A^T B,  A,B: N x {P,Q}
// f32 path: V_WMMA_F32_16X16X4_F32, per-wave 16x16 f32 accumulator tile.
//
// Lanes whose column index exceeds P/Q alias plane 0 and load unconditionally:
// D[r][c] depends only on A-row r and B-col c, so garbage rows/cols of the
// accumulator are never stored and no EXEC masking is needed in the hot loop.
// Addresses advance by pointer increments (one v_add_nc_u64 per stream).
// ---------------------------------------------------------------------------
struct Ptr8 { const float* p[8]; };

template<int ESA,int ESB,int TRA>
__global__ void k_gram_t(Ptr8 A, Ptr8 B, int N, int P, int Q, int lgSide,
                         float invD, float* outBase, int rowStr, int colStr){
  int lane=(int)(threadIdx.x&31);
  int col=lane&15;
  int kb=(lane>>4)<<1;
  const float* pa=A.p[col<P?col:0];
  const float* pb=B.p[col<Q?col:0];
  int wave=(int)(((long)blockIdx.x*blockDim.x+threadIdx.x)>>5);
  int nw=(int)(((long)gridDim.x*blockDim.x)>>5);
  int nch=N>>2;                 // all sizes are multiples of 4
  int side=1<<lgSide;
  int step=nw<<2;               // samples per chunk stride

  auto ld=[&](const float* q,int es)->v2f{
    v2f r;
    if(es==1){ float2 v=*(const float2*)q; r.x=v.x; r.y=v.y; }   // b64
    else     { float4 v=*(const float4*)q; r.x=v.x; r.y=v.z; }   // b128
    return r;
  };

  int n0=(wave<<2)+kb;
  const float* qa=pa+(long)n0*ESA;
  const float* qb=pb+(long)n0*ESB;
  long dA=(long)step*ESA, dB=(long)step*ESB;

  v8f acc0={0.f,0.f,0.f,0.f,0.f,0.f,0.f,0.f};
  v8f acc1={0.f,0.f,0.f,0.f,0.f,0.f,0.f,0.f};
  int c=wave;
  for(; c+nw<nch; c+=nw+nw){
    v2f a0,a1,b0,b1;
    if(TRA){
      int i0=((n0&(side-1))<<lgSide)+(n0>>lgSide);
      int n1=n0+step;
      int i1=((n1&(side-1))<<lgSide)+(n1>>lgSide);
      a0.x=pa[(long)i0*ESA];        a0.y=pa[(long)(i0+side)*ESA];
      a1.x=pa[(long)i1*ESA];        a1.y=pa[(long)(i1+side)*ESA];
      n0+=step<<1;
    }else{
      __builtin_prefetch((const void*)(qa+(dA<<1)),0,1);
      a0=ld(qa,ESA); a1=ld(qa+dA,ESA);
      qa+=dA<<1;
    }
    __builtin_prefetch((const void*)(qb+(dB<<1)),0,1);
    b0=ld(qb,ESB); b1=ld(qb+dB,ESB);
    qb+=dB<<1;
    acc0=__builtin_amdgcn_wmma_f32_16x16x4_f32(false,a0,false,b0,(short)0,acc0,false,false);
    acc1=__builtin_amdgcn_wmma_f32_16x16x4_f32(false,a1,false,b1,(short)0,acc1,false,false);
  }
  if(c<nch){
    v2f a0,b0;
    if(TRA){
      int i0=((n0&(side-1))<<lgSide)+(n0>>lgSide);
      a0.x=pa[(long)i0*ESA]; a0.y=pa[(long)(i0+side)*ESA];
    }else{
      a0=ld(qa,ESA);
    }
    b0=ld(qb,ESB);
    acc0=__builtin_amdgcn_wmma_f32_16x16x4_f32(false,a0,false,b0,(short)0,acc0,false,false);
  }
  acc0=acc0+acc1;
  int rb=(lane>=16)?8:0;
  for(int v=0;v<8;v++){
    int r=v+rb;
    if(r<P && col<Q) atomicAdd(outBase+(long)r*rowStr+(long)col*colStr, acc0[v]*invD);
  }
}

// ---------------------------------------------------------------------------
// host orchestration
// ---------------------------------------------------------------------------
static inline int nblk(long n){ return (int)((n+255)/256); }
static inline int rblk(long n){ long g=(n+255)/256; return (int)(g>1024?1024:g); }
static inline int il2(int v){ int l=0; while((1<<l)<v) l++; return l; }

extern "C" void kernel_launch(void* const* d_in, const int* in_sizes, int n_in,
                              void* d_out_, int out_size, void* d_ws, size_t ws_size,
                              hipStream_t stream){
  (void)in_sizes;(void)n_in;(void)out_size;(void)ws_size;
  const float* im0=(const float*)d_in[0];
  float* out=(float*)d_out_;
  char* ws=(char*)d_ws;

  // workspace layout
  const size_t N0=1048576;            // 1024^2
  const size_t BANDTOT=4ull*(1048576+262144+65536+16384);
  const size_t UNTOT=1048576+262144+65536+16384;
  const size_t OFF_BUFA=1024;
  const size_t OFF_BUFB=OFF_BUFA+8ull*N0;
  const size_t OFF_BUFC=OFF_BUFB+8ull*N0;
  const size_t OFF_HI0 =OFF_BUFC+8ull*N0;
  const size_t OFF_BAND=OFF_HI0+4ull*N0;
  const size_t OFF_AMAG=OFF_BAND+8ull*BANDTOT;
  const size_t OFF_UNOR=OFF_AMAG+4ull*BANDTOT;
  const size_t OFF_LOWR=OFF_UNOR+4ull*UNTOT;
  const size_t OFF_IMLP=OFF_LOWR+4ull*4096;
  const size_t OFF_PAR =OFF_IMLP+4ull*4096;
  const size_t OFF_RPAR=OFF_PAR+4ull*4*N0;

  float*  scal=(float*)ws;            // 0 mean,1 min,2 max,3 s2,4 s3,5 s4,6 var0,7 tmp,8 vari,9 m3,10 m4
  float2* BUFA=(float2*)(ws+OFF_BUFA);
  float2* BUFB=(float2*)(ws+OFF_BUFB);
  float2* BUFC=(float2*)(ws+OFF_BUFC);
  float*  HI0 =(float*)(ws+OFF_HI0);
  float*  LOWR=(float*)(ws+OFF_LOWR);
  float*  IMLP=(float*)(ws+OFF_IMLP);
  float*  PAR =(float*)(ws+OFF_PAR);
  float*  RPAR=(float*)(ws+OFF_RPAR);

  auto bandOff=[&](int s,int b)->size_t{
    size_t o=0; for(int t=0;t<s;t++){ size_t mt=(size_t)(1024>>t); o+=4ull*mt*mt; }
    size_t ms=(size_t)(1024>>s); return o+(size_t)b*ms*ms;
  };
  auto bandp=[&](int s,int b)->float2*{ return (float2*)(ws+OFF_BAND)+bandOff(s,b); };
  auto amagp=[&](int s,int b)->float* { return (float*)(ws+OFF_AMAG)+bandOff(s,b); };
  auto unorp=[&](int s)->float*{
    size_t o=0; for(int t=0;t<s;t++){ size_t mt=(size_t)(1024>>t); o+=mt*mt; }
    return (float*)(ws+OFF_UNOR)+o;
  };

  auto fft2=[&](float2* d,int n,int dir){
    int lg=il2(n);
    float sc=dir>0?1.0f:1.0f/(float)n;
    float ds=dir>0?1.0f:-1.0f;
    k_fft1d<<<n,256,0,stream>>>(d,n,lg,0,ds,sc);
    k_fft1d<<<n,256,0,stream>>>(d,n,lg,1,ds,sc);
  };
  auto zslot=[&](float* p){ k_setf<<<1,1,0,stream>>>(p,1,0.0f); };
  auto rsum=[&](const float* x,long n,int pw,int useAbs,float scale,float* slot){
    zslot(slot);
    k_rsum<<<rblk(n),256,0,stream>>>(x,n,pw,useAbs,scale,slot);
  };
  auto autocorr=[&](const float* x,int m,float denom,float* outBase,int strideIJ,float* variSlot){
    long n2=(long)m*m;
    k_r2c<<<nblk(n2),256,0,stream>>>(x,BUFB,n2);
    fft2(BUFB,m,+1);
    k_powspec<<<nblk(n2),256,0,stream>>>(BUFB,n2);
    fft2(BUFB,m,-1);
    k_acwrite<<<1,128,0,stream>>>(BUFB,m,1.0f/denom,outBase,strideIJ,variSlot);
  };

  // output offsets (floats)
  const int O_STATG=0, O_MM=6, O_ACE=24, O_SK=1320, O_KU=1325, O_ACR=1330,
            O_C0=1735, O_CX0=1815, O_CR0=1879, O_CRX0=2199, O_VHPR=2455;

  // ---- init ----
  k_setf<<<nblk(2456),256,0,stream>>>(out,2456,0.0f);
  k_setf<<<1,256,0,stream>>>(scal,256,0.0f);

  // ---- global stats of im0 ----
  rsum(im0,(long)N0,1,0,1.0f/(float)N0,&scal[0]);
  k_setf<<<1,1,0,stream>>>(&scal[1],1,3.4e38f);
  k_setf<<<1,1,0,stream>>>(&scal[2],1,-3.4e38f);
  k_minmax<<<rblk(N0),256,0,stream>>>(im0,(long)N0,&scal[1],&scal[2]);
  zslot(&scal[3]); zslot(&scal[4]); zslot(&scal[5]);
  k_central<<<rblk(N0),256,0,stream>>>(im0,(long)N0,&scal[0],&scal[3],&scal[4],&scal[5]);
  k_statg<<<1,1,0,stream>>>(scal,out+O_STATG,(float)N0);

  // ---- build pyramid ----
  k_r2c<<<nblk(N0),256,0,stream>>>(im0,BUFA,(long)N0);
  fft2(BUFA,1024,+1);
  k_hi0lo0<<<nblk(N0),256,0,stream>>>(BUFA,BUFB,BUFA,1024,10);  // hi->BUFB, lo->BUFA
  fft2(BUFB,1024,-1);
  k_real<<<nblk(N0),256,0,stream>>>(BUFB,HI0,(long)N0);

  for(int s=0;s<4;s++){
    int m=1024>>s; int lg=10-s; long n2=(long)m*m;
    for(int b=0;b<4;b++){
      k_bandmask<<<nblk(n2),256,0,stream>>>(BUFA,bandp(s,b),m,PI_F*(float)b/4.0f,lg);
      fft2(bandp(s,b),m,-1);
    }
    k_him<<<nblk(n2),256,0,stream>>>(BUFA,BUFB,m,lg);
    fft2(BUFB,m,-1);
    k_real<<<nblk(n2),256,0,stream>>>(BUFB,unorp(s),n2);
    k_croplo<<<nblk((long)(m/2)*(m/2)),256,0,stream>>>(BUFA,BUFC,m,lg-1);
    k_copyc<<<nblk((long)(m/2)*(m/2)),256,0,stream>>>(BUFC,BUFA,(long)(m/2)*(m/2));
  }
  fft2(BUFA,64,-1);
  k_real<<<nblk(4096),256,0,stream>>>(BUFA,LOWR,4096);
  rsum(LOWR,4096,1,0,1.0f/4096.0f,&scal[7]);
  k_subs<<<nblk(4096),256,0,stream>>>(LOWR,4096,&scal[7]);   // lowpass -= mean

  // ---- magMeans + mean-subtracted band magnitudes ----
  rsum(HI0,(long)N0,1,1,1.0f/(float)N0,out+O_MM+0);
  for(int s=0;s<4;s++){
    int m=1024>>s; long n2=(long)m*m;
    for(int b=0;b<4;b++){
      k_mag<<<nblk(n2),256,0,stream>>>(bandp(s,b),amagp(s,b),n2);
      rsum(amagp(s,b),n2,1,0,1.0f/(float)n2,out+O_MM+1+s*4+b);
      k_subs<<<nblk(n2),256,0,stream>>>(amagp(s,b),n2,out+O_MM+1+s*4+b);
    }
  }
  rsum(LOWR,4096,1,1,1.0f/4096.0f,out+O_MM+17);

  // ---- lowpass residual channel (imlp) + its autocorr / skew / kurt ----
  k_r2c<<<nblk(4096),256,0,stream>>>(LOWR,BUFB,4096);
  fft2(BUFB,64,+1);
  k_lo0m<<<nblk(4096),256,0,stream>>>(BUFB,64,6);
  fft2(BUFB,64,-1);
  k_real<<<nblk(4096),256,0,stream>>>(BUFB,IMLP,4096);
  autocorr(IMLP,64,4096.0f,out+O_ACR+4,5,&scal[8]);
  rsum(IMLP,4096,3,0,1.0f/4096.0f,&scal[9]);
  rsum(IMLP,4096,4,0,1.0f/4096.0f,&scal[10]);
  k_skewkurt<<<1,1,0,stream>>>(scal,1.0f,out+O_SK+4,out+O_KU+4);

  // ---- per-scale autocorrs + unoriented skew/kurt ----
  for(int s=0;s<4;s++){
    int m=1024>>s; long n2=(long)m*m;
    for(int b=0;b<4;b++)
      autocorr(amagp(s,b),m,(float)n2,out+O_ACE+s*4+b,16,(float*)0);
    autocorr(unorp(s),m,2.0f*(float)n2,out+O_ACR+s,5,&scal[8]);
    rsum(unorp(s),n2,3,0,1.0f/(float)n2,&scal[9]);
    rsum(unorp(s),n2,4,0,1.0f/(float)n2,&scal[10]);
    k_skewkurt<<<1,1,0,stream>>>(scal,0.5f,out+O_SK+s,out+O_KU+s);
  }

  // ---- correlation matrices (WMMA f32 16x16x4 Gram tiles) ----
  for(int s=0;s<4;s++){
    int m=1024>>s; int lg=10-s; long csz=(long)m*m;
    Ptr8 cous{}; Ptr8 rcous{};
    for(int b=0;b<4;b++){ cous.p[b]=amagp(s,b); rcous.p[b]=(const float*)bandp(s,b); }
    for(int b=4;b<8;b++){ cous.p[b]=amagp(s,0); rcous.p[b]=(const float*)bandp(s,0); }

    // C0[:,:,s] = cous^T cous / csz
    k_gram_t<1,1,0><<<256,256,0,stream>>>(cous,cous,(int)csz,4,4,lg,
                                          1.0f/(float)csz,out+O_C0+s,20,5);

    if(s<3){
      int m2=m>>1; long c2=(long)m2*m2;
      Ptr8 par{}; Ptr8 rpar{};
      for(int b=0;b<4;b++){
        // expand band (s+1,b) in frequency domain: fft -> zeropad -> ifft
        k_copyc<<<nblk(c2),256,0,stream>>>(bandp(s+1,b),BUFB,c2);
        fft2(BUFB,m2,+1);
        k_setf<<<nblk(2*csz),256,0,stream>>>((float*)BUFC,2*csz,0.0f);
        k_zeropad<<<nblk(c2),256,0,stream>>>(BUFB,BUFC,m2,lg-1);
        fft2(BUFC,m,-1);
        k_parents<<<nblk(csz),256,0,stream>>>(BUFC,PAR+(size_t)b*csz,
                                              RPAR+(size_t)b*csz,RPAR+(size_t)(4+b)*csz,csz);
        rsum(PAR+(size_t)b*csz,csz,1,0,1.0f/(float)csz,&scal[7]);
        k_subs<<<nblk(csz),256,0,stream>>>(PAR+(size_t)b*csz,csz,&scal[7]);
        par.p[b]=PAR+(size_t)b*csz;
        rpar.p[b]=RPAR+(size_t)b*csz; rpar.p[4+b]=RPAR+(size_t)(4+b)*csz;
      }
      for(int b=4;b<8;b++) par.p[b]=par.p[0];
      // Cx0 = cous^T parents / csz
      k_gram_t<1,1,0><<<256,256,0,stream>>>(cous,par,(int)csz,4,4,lg,
                                            1.0f/(float)csz,out+O_CX0+s,16,4);
      // Cr0[:4,:4,s] = rcous^T rcous / csz  (real parts of complex bands)
      k_gram_t<2,2,0><<<256,256,0,stream>>>(rcous,rcous,(int)csz,4,4,lg,
                                            1.0f/(float)csz,out+O_CR0+s,40,5);
      // Crx0[:4,:8,s] = rcous^T rparents / csz
      k_gram_t<2,1,0><<<256,256,0,stream>>>(rcous,rpar,(int)csz,4,8,lg,
                                            1.0f/(float)csz,out+O_CRX0+s,32,4);
    } else {
      // last scale: rparents from expanded lowpass (+4 rolled copies)
      k_r2c<<<nblk(4096),256,0,stream>>>(LOWR,BUFB,4096);
      fft2(BUFB,64,+1);
      k_setf<<<nblk(2*csz),256,0,stream>>>((float*)BUFC,2*csz,0.0f);
      k_zeropad<<<nblk(4096),256,0,stream>>>(BUFB,BUFC,64,6);
      fft2(BUFC,m,-1);
      k_lowrp<<<nblk(csz),256,0,stream>>>(BUFC,RPAR,RPAR+csz,RPAR+2*csz,RPAR+3*csz,RPAR+4*csz,m,7);
      Ptr8 rpar{};
      for(int q=0;q<5;q++) rpar.p[q]=RPAR+(size_t)q*csz;
      for(int q=5;q<8;q++) rpar.p[q]=RPAR;
      // Cr0[:4,:4,3] = rcous^T rcous / csz
      k_gram_t<2,2,0><<<256,256,0,stream>>>(rcous,rcous,(int)csz,4,4,lg,
                                            1.0f/(float)csz,out+O_CR0+3,40,5);
      // Crx0[:4,:5,3]: rcous is .T.ravel ordered, rparents is .ravel ordered
      k_gram_t<2,1,1><<<256,256,0,stream>>>(rcous,rpar,(int)csz,4,5,lg,
                                            1.0f/(float)csz,out+O_CRX0+3,32,4);
      // Cr0[:5,:5,4] = rparents^T rparents / (csz/4)
      k_gram_t<1,1,0><<<256,256,0,stream>>>(rpar,rpar,(int)csz,5,5,lg,
                                            4.0f/(float)csz,out+O_CR0+4,40,5);
    }
  }

  // ---- highpass residual variance ----
  rsum(HI0,(long)N0,2,0,1.0f/(float)N0,out+O_VHPR);
}